// MultiScaleTemporalConv_36971078484447
// MI455X (gfx1250) — compile-verified
//
#include <hip/hip_runtime.h>
#include <hip/hip_bf16.h>
#include <math.h>

// Problem constants (from reference)
#define BB   4
#define TT   256
#define NN   2048
#define CIN  32
#define COUT 48
#define EK   64
#define ECL  62
#define EE   126          // EK + ECL
#define MAXK 12
#define PP   6            // MAX_KERNEL/2
#define LL   13           // 2P+1
#define KTOT 416          // LL * CIN
#define WSZ  (KTOT*COUT)  // 19968 floats per entity

typedef __attribute__((ext_vector_type(2))) float v2f;
typedef __attribute__((ext_vector_type(8))) float v8f;

// ---------------------------------------------------------------------------
// Kernel 1: condensed[b,t,e,c] = keys / cluster means
// ---------------------------------------------------------------------------
__global__ __launch_bounds__(256) void condense_kernel(
    const float* __restrict__ x, const int* __restrict__ keyIdx,
    const int* __restrict__ cm, float* __restrict__ condensed)
{
    const int bt = blockIdx.x;            // 0..1023  (= b*T + t)
    const int c  = threadIdx.x & 31;
    const int eg = threadIdx.x >> 5;      // 0..7
    const float* xbt = x + (size_t)bt * NN * CIN;
    float* cbt = condensed + (size_t)bt * EE * CIN;

    for (int e = eg; e < EK; e += 8)
        cbt[e * CIN + c] = xbt[(size_t)keyIdx[e] * CIN + c];

    for (int ec = eg; ec < ECL; ec += 8) {
        float s = 0.0f;
        #pragma unroll 4
        for (int j = 0; j < 32; ++j)
            s += xbt[(size_t)cm[ec * 32 + j] * CIN + c];
        cbt[(EK + ec) * CIN + c] = s * (1.0f / 32.0f);
    }
}

// ---------------------------------------------------------------------------
// Kernel 2: per-entity kernel-size estimation (pooled mean/var + DFT argmax)
// ---------------------------------------------------------------------------
__global__ __launch_bounds__(256) void estimate_kernel(
    const float* __restrict__ condensed, int* __restrict__ kl_out,
    int* __restrict__ ks_out)
{
    const int e = blockIdx.x;     // 0..125
    const int t = threadIdx.x;    // 0..255
    __shared__ float sp[TT];      // pooled, then centered series
    __shared__ float sr[TT];      // reduction / amplitudes
    __shared__ float s_var;

    float s = 0.0f;
    for (int b = 0; b < BB; ++b) {
        const float* p = condensed + (((size_t)(b * TT + t) * EE + e) << 5);
        #pragma unroll 8
        for (int c = 0; c < CIN; ++c) s += p[c];
    }
    sp[t] = s * (1.0f / (BB * CIN));
    __syncthreads();

    sr[t] = sp[t];
    __syncthreads();
    for (int ofs = 128; ofs > 0; ofs >>= 1) {
        if (t < ofs) sr[t] += sr[t + ofs];
        __syncthreads();
    }
    float mean = sr[0] * (1.0f / TT);
    __syncthreads();

    float cen = sp[t] - mean;
    sp[t] = cen;
    sr[t] = cen * cen;
    __syncthreads();
    for (int ofs = 128; ofs > 0; ofs >>= 1) {
        if (t < ofs) sr[t] += sr[t + ofs];
        __syncthreads();
    }
    if (t == 0) s_var = sr[0] * (1.0f / TT);
    __syncthreads();

    // DFT amplitude for frequency f = t (valid bins 1..128)
    float amp = -1.0f;
    if (t >= 1 && t <= 128) {
        float re = 0.0f, im = 0.0f;
        const float w = -2.0f * 3.14159265358979323846f * (float)t / (float)TT;
        for (int tt2 = 0; tt2 < TT; ++tt2) {
            float ph = w * (float)tt2;
            re += sp[tt2] * cosf(ph);
            im += sp[tt2] * sinf(ph);
        }
        amp = sqrtf(re * re + im * im);
    }
    sr[t] = amp;
    __syncthreads();

    if (t == 0) {
        int best = 0; float bamp = 0.0f;     // amp[0] forced to 0 in reference
        for (int f = 1; f <= 128; ++f)
            if (sr[f] > bamp) { bamp = sr[f]; best = f; }
        int kl_full;
        if (best <= 0) kl_full = 128;
        else {
            kl_full = (int)((float)TT / (float)best);   // truncation, like astype(int32)
            if (kl_full < 2) kl_full = 2;
        }
        if (kl_full > 128) kl_full = 128;

        float scale = 1.0f + log1pf(s_var * 1.0e5f);    // var / EPS, EPS=1e-5
        float sden = scale > 1.0f ? scale : 1.0f;
        int ks_full = (int)rintf((float)kl_full / sden); // round-half-even like jnp.round
        if (ks_full < 2) ks_full = 2;
        if (ks_full > 128) ks_full = 128;

        kl_out[e] = kl_full > MAXK ? MAXK : kl_full;    // outer clip to [1,12]
        ks_out[e] = ks_full > MAXK ? MAXK : ks_full;
    }
}

// ---------------------------------------------------------------------------
// Kernel 3: per-entity fused (gate-folded) conv weight, layout Wf[e][K=m*32+i][o]
// ---------------------------------------------------------------------------
__global__ __launch_bounds__(256) void weight_prep_kernel(
    const float* __restrict__ LW, const float* __restrict__ SW,
    const float* __restrict__ LB, const float* __restrict__ SB,
    const float* __restrict__ gate, const int* __restrict__ klA,
    const int* __restrict__ ksA, float* __restrict__ Wf, float* __restrict__ bf)
{
    const int e  = blockIdx.x;
    const int kl = klA[e], ks = ksA[e];
    const int pl = kl >> 1, ps = ks >> 1;

    for (int idx = threadIdx.x; idx < WSZ; idx += 256) {
        int o = idx % COUT;
        int K = idx / COUT;
        int m = K >> 5;       // tap 0..12
        int i = K & 31;       // input channel
        float g = 1.0f / (1.0f + expf(-gate[o]));

        int jl = m - PP + pl;
        float wl = 0.0f;
        if (jl >= 0 && jl < kl) {
            int jc = jl > (MAXK - 1) ? (MAXK - 1) : jl;
            wl = LW[(o * CIN + i) * MAXK + jc];
        }
        int js = m - PP + ps;
        float wsv = 0.0f;
        if (js >= 0 && js < ks) {
            int jc = js > (MAXK - 1) ? (MAXK - 1) : js;
            wsv = SW[(o * CIN + i) * MAXK + jc];
        }
        Wf[(size_t)e * WSZ + idx] = g * wl + (1.0f - g) * wsv;
    }
    if (e == 0 && threadIdx.x < COUT) {
        int o = threadIdx.x;
        float g = 1.0f / (1.0f + expf(-gate[o]));
        bf[o] = g * LB[o] + (1.0f - g) * SB[o];
    }
}

// ---------------------------------------------------------------------------
// Kernel 4: WMMA conv (D = A(16xK) x Wf(Kx48) + bias) fused with node scatter.
// 8 blocks per entity (blockIdx.y); Wf[e] staged in LDS (416x48 f32 = 80KB); 8 waves each
// own one 16-row (b,t) tile; K consumed 4 at a time via v_wmma_f32_16x16x4_f32.
// ---------------------------------------------------------------------------
__global__ __launch_bounds__(256) void conv_scatter_kernel(
    const float* __restrict__ condensed, const float* __restrict__ Wf,
    const float* __restrict__ bf, const int* __restrict__ keyIdx,
    const int* __restrict__ cm, float* __restrict__ out)
{
    const int e = blockIdx.x;                 // entity
    __shared__ float sW[WSZ];                 // 79,872 bytes
    const float* gW = Wf + (size_t)e * WSZ;
    for (int i = threadIdx.x; i < WSZ; i += 256) sW[i] = gW[i];
    __syncthreads();

    const int lane = threadIdx.x & 31;
    const int wave = threadIdx.x >> 5;
    const int mt   = blockIdx.y * 8 + wave;   // 0..63  (M = B*T = 1024 rows)
    const int R    = mt * 16;
    const int b    = R >> 8;                  // tile never crosses batch (256%16==0)
    const int t0   = R & 255;

    const int colN = lane & 15;               // N (and A-row) index within tile
    const int hi   = lane >> 4;               // lanes 16-31 carry K+2
    const int klo  = hi * 2;
    const int tA   = t0 + colN;               // A row's base time

    v8f acc0 = {}; v8f acc1 = {}; v8f acc2 = {};

    for (int m = 0; m < LL; ++m) {
        int ts = tA + m - PP;
        float mask = (ts >= 0 && ts < TT) ? 1.0f : 0.0f;   // lane-select, no branch
        int tc = ts < 0 ? 0 : (ts > TT - 1 ? TT - 1 : ts);
        const float* xrow = condensed + (((size_t)(b * TT + tc) * EE + e) << 5);
        #pragma unroll
        for (int ic = 0; ic < 8; ++ic) {
            const int K = m * 32 + ic * 4;
            v2f a;
            a.x = xrow[ic * 4 + klo]     * mask;   // A: K = klo
            a.y = xrow[ic * 4 + klo + 1] * mask;   // A: K = klo+1
            const float* wrow = &sW[(K + klo) * COUT];
            v2f b0, b1, b2;
            b0.x = wrow[colN];          b0.y = wrow[COUT + colN];
            b1.x = wrow[16 + colN];     b1.y = wrow[COUT + 16 + colN];
            b2.x = wrow[32 + colN];     b2.y = wrow[COUT + 32 + colN];
            acc0 = __builtin_amdgcn_wmma_f32_16x16x4_f32(false, a, false, b0, (short)0, acc0, false, false);
            acc1 = __builtin_amdgcn_wmma_f32_16x16x4_f32(false, a, false, b1, (short)0, acc1, false, false);
            acc2 = __builtin_amdgcn_wmma_f32_16x16x4_f32(false, a, false, b2, (short)0, acc2, false, false);
        }
    }

    const float bias0 = bf[colN];
    const float bias1 = bf[16 + colN];
    const float bias2 = bf[32 + colN];

    const int nNodes = (e < EK) ? 1 : 32;
    for (int j = 0; j < nNodes; ++j) {
        const int node = (e < EK) ? keyIdx[e] : cm[(e - EK) * 32 + j];
        #pragma unroll
        for (int v = 0; v < 8; ++v) {
            const int t = t0 + v + hi * 8;     // C/D layout: VGPR v -> M=v (+8 hi)
            float* orow = out + (((size_t)(b * TT + t) * NN) + node) * COUT;
            orow[colN]      = acc0[v] + bias0;
            orow[16 + colN] = acc1[v] + bias1;
            orow[32 + colN] = acc2[v] + bias2;
        }
    }
}

// ---------------------------------------------------------------------------
extern "C" void kernel_launch(void* const* d_in, const int* in_sizes, int n_in,
                              void* d_out, int out_size, void* d_ws, size_t ws_size,
                              hipStream_t stream) {
    const float* x      = (const float*)d_in[0];
    const int*   keyIdx = (const int*)d_in[1];
    const int*   cm     = (const int*)d_in[2];
    const float* LW     = (const float*)d_in[3];
    const float* SW     = (const float*)d_in[4];
    const float* LB     = (const float*)d_in[5];
    const float* SB     = (const float*)d_in[6];
    const float* gate   = (const float*)d_in[7];
    float* out = (float*)d_out;

    // Workspace layout (floats): condensed | Wf | bf | kl | ks  -> ~26.6 MB
    float* ws        = (float*)d_ws;
    float* condensed = ws;                                   // 4,128,768
    float* Wf        = condensed + (size_t)BB * TT * EE * CIN; // 2,515,968
    float* bf        = Wf + (size_t)EE * WSZ;                // 48 (pad 64)
    int*   klA       = (int*)(bf + 64);                      // 126 (pad 128)
    int*   ksA       = klA + 128;

    condense_kernel<<<BB * TT, 256, 0, stream>>>(x, keyIdx, cm, condensed);
    estimate_kernel<<<EE, 256, 0, stream>>>(condensed, klA, ksA);
    weight_prep_kernel<<<EE, 256, 0, stream>>>(LW, SW, LB, SB, gate, klA, ksA, Wf, bf);
    conv_scatter_kernel<<<dim3(EE, 8), 256, 0, stream>>>(condensed, Wf, bf, keyIdx, cm, out);
}